// SimpleRNN_17600775979833
// MI455X (gfx1250) — compile-verified
//
#include <hip/hip_runtime.h>

#define VOCAB  32000
#define EMBED  256
#define HIDDEN 512
#define BATCH  16
#define SEQ    512

typedef __attribute__((ext_vector_type(2))) float v2f;
typedef __attribute__((ext_vector_type(8))) float v8f;

__device__ __forceinline__ v8f wmma_f32(v2f a, v2f b, v8f c) {
    // V_WMMA_F32_16X16X4_F32: D = A(16x4) x B(4x16) + C(16x16), fp32 matrix pipe
    return __builtin_amdgcn_wmma_f32_16x16x4_f32(
        /*neg_a=*/false, a, /*neg_b=*/false, b,
        /*c_mod=*/(short)0, c, /*reuse_a=*/false, /*reuse_b=*/false);
}

// ---------------------------------------------------------------------------
// Kernel A: x_proj[r, n] = sum_k emb[x[r], k] * W_ih[n, k] + b_ih[n] + b_hh[n]
// r = b*SEQ + t in [0, 8192). Embedding gather folded into the A-fragment load.
// Block = 256 threads = 8 waves; block owns one M-tile (16 rows) x full N=512,
// each wave owns 4 N-tiles (64 cols) so one A-fragment feeds 4 WMMAs.
// ---------------------------------------------------------------------------
__global__ __launch_bounds__(256) void xproj_kernel(
    const int* __restrict__ x, const float* __restrict__ emb,
    const float* __restrict__ W_ih, const float* __restrict__ b_ih,
    const float* __restrict__ b_hh, float* __restrict__ xp) {
    const int lane    = threadIdx.x & 31;
    const int wave    = threadIdx.x >> 5;
    const int l15     = lane & 15;
    const int kh      = (lane >> 4) << 1;   // 0 or 2
    const int mrow    = (lane >> 4) << 3;   // 0 or 8
    const int rowBase = blockIdx.x * 16;
    const int colBase = wave * 64;

    const int    idx  = x[rowBase + l15];               // gather index, row m=l15
    const float* arow = emb + (long)idx * EMBED;

    v8f acc[4];
#pragma unroll
    for (int j = 0; j < 4; ++j) {
        const float bias = b_ih[colBase + j * 16 + l15] + b_hh[colBase + j * 16 + l15];
#pragma unroll
        for (int v = 0; v < 8; ++v) acc[j][v] = bias;
    }

    for (int kk = 0; kk < EMBED / 4; ++kk) {
        const int k0 = kk * 4 + kh;
        const v2f a  = *(const v2f*)(arow + k0);
#pragma unroll
        for (int j = 0; j < 4; ++j) {
            const int n = colBase + j * 16 + l15;
            const v2f b = *(const v2f*)(W_ih + (long)n * EMBED + k0);
            acc[j] = wmma_f32(a, b, acc[j]);
        }
    }

#pragma unroll
    for (int j = 0; j < 4; ++j) {
        const int n = colBase + j * 16 + l15;
#pragma unroll
        for (int v = 0; v < 8; ++v)
            xp[(long)(rowBase + v + mrow) * HIDDEN + n] = acc[j][v];
    }
}

// ---------------------------------------------------------------------------
// Kernel B: serial recurrence. h double-buffered in LDS (64 KB of 320 KB WGP
// LDS). One workgroup, 1024 threads = 32 waves; wave w owns N-tile [16w,16w+16).
// Per step: acc = x_proj[:, t, tile] ; 128 x WMMA over K=512 ; tanh ; write
// h_new to the other LDS buffer AND back over x_proj (becomes `out`).
// W_hh (1 MB) stays L2-resident across all 512 steps.
// ---------------------------------------------------------------------------
__global__ __launch_bounds__(1024) void rnn_kernel(
    const float* __restrict__ hidden, const float* __restrict__ W_hh,
    float* __restrict__ xp, float* __restrict__ h_last) {
    __shared__ float hbuf[2 * BATCH * HIDDEN];   // 2 x 16 x 512 f32 = 64 KB

    const int tid  = threadIdx.x;
    const int lane = tid & 31;
    const int wave = tid >> 5;             // N-tile id, 0..31
    const int l15  = lane & 15;
    const int kh   = (lane >> 4) << 1;
    const int mrow = (lane >> 4) << 3;
    const int col  = wave * 16 + l15;

    for (int i = tid; i < BATCH * HIDDEN; i += 1024) hbuf[i] = hidden[i];
    __syncthreads();

    int p = 0;
    for (int t = 0; t < SEQ; ++t) {
        v8f acc;
#pragma unroll
        for (int v = 0; v < 8; ++v)
            acc[v] = xp[((long)(v + mrow) * SEQ + t) * HIDDEN + col];

        const float* hrd  = hbuf + p * (BATCH * HIDDEN) + l15 * HIDDEN;
        const float* wrow = W_hh + (long)col * HIDDEN;
#pragma unroll 4
        for (int kk = 0; kk < HIDDEN / 4; ++kk) {
            const int k0 = kk * 4 + kh;
            const v2f a = *(const v2f*)(hrd + k0);
            const v2f b = *(const v2f*)(wrow + k0);
            acc = wmma_f32(a, b, acc);
        }

        float* hwr = hbuf + (p ^ 1) * (BATCH * HIDDEN);
#pragma unroll
        for (int v = 0; v < 8; ++v) {
            const int m = v + mrow;
            const float f = tanhf(acc[v]);
            hwr[m * HIDDEN + col] = f;
            // `outs` row for the final GEMM; read once, much later -> NT store
            __builtin_nontemporal_store(f, &xp[((long)m * SEQ + t) * HIDDEN + col]);
        }
        __syncthreads();
        p ^= 1;
    }

    const float* hf = hbuf + p * (BATCH * HIDDEN);
    for (int i = tid; i < BATCH * HIDDEN; i += 1024) h_last[i] = hf[i];
}

// ---------------------------------------------------------------------------
// Kernel C: logits = out @ fc_w^T + fc_b.  M=8192, N=32000, K=512.
// Store-bandwidth-bound (1.05 GB out / 23.3 TB/s ~ 45 us floor), so fp32 WMMA
// is the right precision. Output is write-once/never-read: store with TH=NT so
// the 1 GB stream doesn't evict L2-resident fc_w (64 MB) + out (16 MB).
// Block = 256 threads / 8 waves, wave owns 2 N-tiles (32 cols) -> block covers
// 16 x 256; grid = (32000/256, 8192/16).
// ---------------------------------------------------------------------------
__global__ __launch_bounds__(256) void logits_kernel(
    const float* __restrict__ out, const float* __restrict__ fc_w,
    const float* __restrict__ fc_b, float* __restrict__ logits) {
    const int lane    = threadIdx.x & 31;
    const int wave    = threadIdx.x >> 5;
    const int l15     = lane & 15;
    const int kh      = (lane >> 4) << 1;
    const int mrow    = (lane >> 4) << 3;
    const int rowBase = blockIdx.y * 16;
    const int colBase = blockIdx.x * 256 + wave * 32;

    v8f acc[2];
#pragma unroll
    for (int j = 0; j < 2; ++j) {
        const float bias = fc_b[colBase + j * 16 + l15];
#pragma unroll
        for (int v = 0; v < 8; ++v) acc[j][v] = bias;
    }

    const float* arow = out + (long)(rowBase + l15) * HIDDEN;
#pragma unroll 4
    for (int kk = 0; kk < HIDDEN / 4; ++kk) {
        const int k0 = kk * 4 + kh;
        const v2f a = *(const v2f*)(arow + k0);
#pragma unroll
        for (int j = 0; j < 2; ++j) {
            const int n = colBase + j * 16 + l15;
            const v2f b = *(const v2f*)(fc_w + (long)n * HIDDEN + k0);
            acc[j] = wmma_f32(a, b, acc[j]);
        }
    }

#pragma unroll
    for (int j = 0; j < 2; ++j) {
        const int n = colBase + j * 16 + l15;
#pragma unroll
        for (int v = 0; v < 8; ++v)
            __builtin_nontemporal_store(
                acc[j][v], &logits[(long)(rowBase + v + mrow) * VOCAB + n]);
    }
}

extern "C" void kernel_launch(void* const* d_in, const int* in_sizes, int n_in,
                              void* d_out, int out_size, void* d_ws, size_t ws_size,
                              hipStream_t stream) {
    const int*   x      = (const int*)d_in[0];
    const float* hidden = (const float*)d_in[1];
    const float* emb    = (const float*)d_in[2];
    const float* W_ih   = (const float*)d_in[3];
    const float* W_hh   = (const float*)d_in[4];
    const float* b_ih   = (const float*)d_in[5];
    const float* b_hh   = (const float*)d_in[6];
    const float* fc_w   = (const float*)d_in[7];
    const float* fc_b   = (const float*)d_in[8];

    float* logits = (float*)d_out;
    float* h_last = logits + (long)BATCH * SEQ * VOCAB;   // tail of d_out
    float* xp     = (float*)d_ws;                         // 16 MB: x_proj, then outs

    // 1) gather + input projection GEMM (8192 x 512 x 256)
    xproj_kernel<<<dim3((BATCH * SEQ) / 16), 256, 0, stream>>>(
        x, emb, W_ih, b_ih, b_hh, xp);

    // 2) serial recurrence: one WGP, LDS-resident double-buffered h
    rnn_kernel<<<1, 1024, 0, stream>>>(hidden, W_hh, xp, h_last);

    // 3) output projection GEMM (8192 x 32000 x 512) — HBM-store bound, NT stores
    logits_kernel<<<dim3(VOCAB / 256, (BATCH * SEQ) / 16), 256, 0, stream>>>(
        xp, fc_w, fc_b, logits);
}